// GCNModel_43018392437092
// MI455X (gfx1250) — compile-verified
//
#include <hip/hip_runtime.h>

// GCN 2-layer forward for MI455X (gfx1250, wave32).
// GEMM1 uses V_WMMA_F32_16X16X4_F32 (exact fp32, matches reference).
// Aggregation uses hardware global_atomic_add_f32; h fits in 192MB L2 so the
// random gather/scatter traffic (~2GB) is L2-resident.

#define NN   100000
#define NE   1600000
#define INC  128
#define HIDC 128
#define OUTC 18

typedef float v2f __attribute__((ext_vector_type(2)));
typedef float v8f __attribute__((ext_vector_type(8)));

__device__ __forceinline__ void atomAddF(float* p, float v) {
  // hardware global_atomic_add_f32 (no CAS loop)
  unsafeAtomicAdd(p, v);
}

// ---- degree / normalization -------------------------------------------------

__global__ void k_init_deg(float* deg) {
  int i = blockIdx.x * blockDim.x + threadIdx.x;
  if (i < NN) deg[i] = 1.0f;               // self-loop contribution
}

__global__ void k_deg(const long long* __restrict__ ei, float* deg) {
  int e = blockIdx.x * blockDim.x + threadIdx.x;
  if (e < NE) {
    int d = (int)ei[NE + e];               // dst row of edge_index
    atomAddF(&deg[d], 1.0f);
  }
}

__global__ void k_dinv(const float* __restrict__ deg, float* __restrict__ dinv) {
  int i = blockIdx.x * blockDim.x + threadIdx.x;
  if (i < NN) dinv[i] = rsqrtf(deg[i]);    // deg >= 1 always (self-loops)
}

// ---- GEMM1: h = x @ W1 via f32 WMMA ----------------------------------------
// One wave per 16x16 output tile. Block = 8 waves = the 8 column tiles of W1.
// grid.x = 100000/16 = 6250 row tiles (exact).

__global__ __launch_bounds__(256) void k_gemm1(const float* __restrict__ x,
                                               const float* __restrict__ W1,
                                               float* __restrict__ h) {
  const int wave = threadIdx.x >> 5;       // n-tile 0..7
  const int lane = threadIdx.x & 31;
  const int r    = lane & 15;
  const int g    = lane >> 4;              // lane group (K split for A/B frags)
  const int row0 = blockIdx.x * 16;
  const int col  = wave * 16 + r;

  v8f c = {};
  const float* xrow = x  + (row0 + r) * INC + 2 * g;   // A[r][4k+2g+v]
  const float* wcol = W1 + (2 * g) * HIDC + col;       // B[4k+2g+v][col]

#pragma unroll
  for (int k = 0; k < 32; ++k) {           // K = 128 = 32 steps of 4
    v2f a = *(const v2f*)(xrow + 4 * k);   // 8B aligned (4k+2g even)
    v2f b;
    b.x = wcol[(4 * k + 0) * HIDC];
    b.y = wcol[(4 * k + 1) * HIDC];
    c = __builtin_amdgcn_wmma_f32_16x16x4_f32(
        /*neg_a=*/false, a, /*neg_b=*/false, b,
        /*c_mod=*/(short)0, c, /*reuse_a=*/false, /*reuse_b=*/false);
  }

#pragma unroll
  for (int j = 0; j < 8; ++j)              // D: VGPR j -> row j+8g, lane r -> col
    h[(row0 + j + 8 * g) * HIDC + col] = c[j];
}

// ---- layer-1 aggregation ----------------------------------------------------
// init: agg1[i] = h[i] * dinv[i]^2   (self-loop folded in, also zeroes buffer)

__global__ void k_selfinit1(const float* __restrict__ h,
                            const float* __restrict__ dinv,
                            float* __restrict__ agg) {
  int gid = blockIdx.x * blockDim.x + threadIdx.x;   // NN*32 threads (float4 each)
  if (gid >= NN * 32) return;
  int i = gid >> 5, q = gid & 31;
  float w = dinv[i] * dinv[i];
  float4 v = ((const float4*)(h + i * HIDC))[q];
  v.x *= w; v.y *= w; v.z *= w; v.w *= w;
  ((float4*)(agg + i * HIDC))[q] = v;
}

__global__ void k_scatter1(const long long* __restrict__ ei,
                           const float* __restrict__ dinv,
                           const float* __restrict__ h,
                           float* __restrict__ agg) {
  int gid = blockIdx.x * blockDim.x + threadIdx.x;   // NE*32 threads
  if (gid >= NE * 32) return;
  int e = gid >> 5, q = gid & 31;                    // wave per edge, float4/lane
  int s = (int)ei[e];
  int d = (int)ei[NE + e];
  float w = dinv[s] * dinv[d];
  float4 v = ((const float4*)(h + s * HIDC))[q];     // coalesced 512B/edge gather
  float* o = agg + d * HIDC + q * 4;
  atomAddF(o + 0, v.x * w);
  atomAddF(o + 1, v.y * w);
  atomAddF(o + 2, v.z * w);
  atomAddF(o + 3, v.w * w);
}

__global__ void k_relu_bias1(float* __restrict__ agg, const float* __restrict__ b1) {
  int gid = blockIdx.x * blockDim.x + threadIdx.x;   // NN*HIDC
  if (gid >= NN * HIDC) return;
  int c = gid & (HIDC - 1);
  agg[gid] = fmaxf(agg[gid] + b1[c], 0.0f);
}

// ---- GEMM2: h2 = h1 @ W2 (128 -> 18), W2 staged in LDS, h1 read once -------

__global__ __launch_bounds__(64) void k_gemm2(const float* __restrict__ h1,
                                              const float* __restrict__ W2,
                                              float* __restrict__ h2) {
  __shared__ float W2s[HIDC * OUTC];
  for (int i = threadIdx.x; i < HIDC * OUTC; i += 64) W2s[i] = W2[i];
  __syncthreads();
  int n = blockIdx.x * 64 + threadIdx.x;
  if (n >= NN) return;
  float acc[OUTC];
#pragma unroll
  for (int c = 0; c < OUTC; ++c) acc[c] = 0.0f;
  const float* row = h1 + n * HIDC;
  for (int k = 0; k < HIDC; ++k) {
    float hk = row[k];
#pragma unroll
    for (int c = 0; c < OUTC; ++c) acc[c] += hk * W2s[k * OUTC + c];  // LDS broadcast
  }
  float* o = h2 + n * OUTC;
#pragma unroll
  for (int c = 0; c < OUTC; ++c) o[c] = acc[c];
}

// ---- layer-2 aggregation ----------------------------------------------------

__global__ void k_selfinit2(const float* __restrict__ h2,
                            const float* __restrict__ dinv,
                            float* __restrict__ agg) {
  int gid = blockIdx.x * blockDim.x + threadIdx.x;   // NN*32 threads
  if (gid >= NN * 32) return;
  int i = gid >> 5, c = gid & 31;
  if (c >= OUTC) return;
  float w = dinv[i] * dinv[i];
  agg[i * OUTC + c] = h2[i * OUTC + c] * w;
}

__global__ void k_scatter2(const long long* __restrict__ ei,
                           const float* __restrict__ dinv,
                           const float* __restrict__ h2,
                           float* __restrict__ agg) {
  int gid = blockIdx.x * blockDim.x + threadIdx.x;   // NE*32 threads
  if (gid >= NE * 32) return;
  int e = gid >> 5, c = gid & 31;
  if (c >= OUTC) return;
  int s = (int)ei[e];
  int d = (int)ei[NE + e];
  float w = dinv[s] * dinv[d];
  atomAddF(&agg[d * OUTC + c], h2[s * OUTC + c] * w);
}

// ---- bias + log_softmax -----------------------------------------------------

__global__ void k_lsm(const float* __restrict__ agg,
                      const float* __restrict__ b2,
                      float* __restrict__ out) {
  int i = blockIdx.x * blockDim.x + threadIdx.x;
  if (i >= NN) return;
  float v[OUTC];
  float m = -3.0e38f;
#pragma unroll
  for (int c = 0; c < OUTC; ++c) {
    v[c] = agg[i * OUTC + c] + b2[c];
    m = fmaxf(m, v[c]);
  }
  float s = 0.0f;
#pragma unroll
  for (int c = 0; c < OUTC; ++c) s += expf(v[c] - m);
  float l = m + logf(s);
#pragma unroll
  for (int c = 0; c < OUTC; ++c) out[i * OUTC + c] = v[c] - l;
}

// ---- host-side orchestration ------------------------------------------------

extern "C" void kernel_launch(void* const* d_in, const int* in_sizes, int n_in,
                              void* d_out, int out_size, void* d_ws, size_t ws_size,
                              hipStream_t stream) {
  const float*     x  = (const float*)d_in[0];
  const long long* ei = (const long long*)d_in[1];   // edge_index is int64
  const float*     W1 = (const float*)d_in[2];
  const float*     b1 = (const float*)d_in[3];
  const float*     W2 = (const float*)d_in[4];
  const float*     b2 = (const float*)d_in[5];
  float*           out = (float*)d_out;

  // workspace layout (floats): deg | dinv | hbuf(N*128) | agg1(N*128)
  // hbuf region is recycled for h2 (N*18) and agg2 (N*18) after scatter1.
  float* ws   = (float*)d_ws;
  float* deg  = ws;
  float* dinv = ws + NN;
  float* hbuf = ws + 2 * NN;
  float* agg1 = ws + 2 * NN + NN * HIDC;
  float* h2   = hbuf;
  float* agg2 = hbuf + NN * OUTC;

  const int B = 256;
  k_init_deg <<<(NN + B - 1) / B, B, 0, stream>>>(deg);
  k_deg      <<<(NE + B - 1) / B, B, 0, stream>>>(ei, deg);
  k_dinv     <<<(NN + B - 1) / B, B, 0, stream>>>(deg, dinv);

  k_gemm1    <<<NN / 16, 256, 0, stream>>>(x, W1, hbuf);            // 6250 blocks

  k_selfinit1<<<(NN * 32 + B - 1) / B, B, 0, stream>>>(hbuf, dinv, agg1);
  k_scatter1 <<<(NE * 32 + B - 1) / B, B, 0, stream>>>(ei, dinv, hbuf, agg1);
  k_relu_bias1<<<(NN * HIDC + B - 1) / B, B, 0, stream>>>(agg1, b1);

  k_gemm2    <<<(NN + 63) / 64, 64, 0, stream>>>(agg1, W2, h2);

  k_selfinit2<<<(NN * 32 + B - 1) / B, B, 0, stream>>>(h2, dinv, agg2);
  k_scatter2 <<<(NE * 32 + B - 1) / B, B, 0, stream>>>(ei, dinv, h2, agg2);

  k_lsm      <<<(NN + B - 1) / B, B, 0, stream>>>(agg2, b2, out);
}